// Lorenz96_torch_9010841387207
// MI455X (gfx1250) — compile-verified
//
#include <hip/hip_runtime.h>
#include <stdint.h>

// Lorenz96 RK4 step, fp32, BATCH x 40 state.
//   dxdt_i = (x_{i+1} - x_{i-2}) * x_{i-1} - x_i + 8
//   x_out  = x + dt/6 * (k1 + 2 k2 + 2 k3 + k4)
//
// Async global<->LDS staging (CDNA5 ASYNCcnt path, zero per-iter addressing),
// per-thread row in registers. Compute uses packed f32 (v_pk_fma_f32 etc.):
// elements are split-interleaved as pairs P[j] = (x_j, x_{j+20}) so that the
// cyclic shifts of the stencil stay pair-aligned; ring wraparound becomes a
// component swap (foldable into VOP3P op_sel), only 4 per dxdt stage.

#define NS 40
#define NH 20                         // packed pairs per row
#define RPB 256                       // rows per block == threads per block
#define NIT 10                        // 16B chunks per thread

typedef float v2f __attribute__((ext_vector_type(2)));

// Packed cyclic shift by d in split-interleave space:
// element pair at j after shift d is (x_{(j+d)%40}, x_{(j+20+d)%40}).
__device__ __forceinline__ v2f sh(const v2f* P, int j, int d) {
    int idx = (j + d + NS) % NS;
    if (idx < NH) return P[idx];
    v2f t = P[idx - NH];
    return __builtin_shufflevector(t, t, 1, 0);   // swap halves (op_sel)
}

__global__ __launch_bounds__(RPB) void lorenz96_rk4_kernel(
    const float* __restrict__ x0,
    const float* __restrict__ dtp,
    float* __restrict__ out)
{
    __shared__ __align__(16) float tile[RPB * NS];   // 40960 B, contiguous rows

    const int tid = threadIdx.x;
    const float dt = dtp[0];

    const float* gin  = x0  + (size_t)blockIdx.x * (RPB * NS);
    float*       gout = out + (size_t)blockIdx.x * (RPB * NS);

    const uint32_t lds_base = (uint32_t)(uintptr_t)tile;
    const uint32_t voff = (uint32_t)tid * 16u;    // global byte offset VGPR
    const uint32_t loff = lds_base + voff;        // LDS  byte address VGPR

    // ---- Stage 1: coalesced async copy global -> LDS ----
    #pragma unroll
    for (int it = 0; it < NIT; ++it) {
        asm volatile("global_load_async_to_lds_b128 %0, %1, %2 offset:%3"
                     :: "v"(loff), "v"(voff), "s"(gin), "i"(it * RPB * 16)
                     : "memory");
    }
    asm volatile("s_wait_asynccnt 0" ::: "memory");
    __syncthreads();

    // ---- Stage 2: read own row (10x ds_load_b128), split-interleave pack ----
    float4* tile4 = (float4*)tile;
    float xs[NS];
    #pragma unroll
    for (int j = 0; j < NS / 4; ++j) {
        float4 v = tile4[tid * (NS / 4) + j];
        xs[4 * j + 0] = v.x; xs[4 * j + 1] = v.y;
        xs[4 * j + 2] = v.z; xs[4 * j + 3] = v.w;
    }

    v2f X[NH], S[NH], K[NH], A[NH];
    #pragma unroll
    for (int j = 0; j < NH; ++j) X[j] = (v2f){xs[j], xs[j + NH]};

    const v2f C8 = (v2f)(8.0f);
    const v2f H2 = (v2f)(0.5f * dt);
    const v2f C6 = (v2f)(dt * (1.0f / 6.0f));
    const v2f C3 = (v2f)(dt * (1.0f / 3.0f));
    const v2f DT = (v2f)(dt);

    // Packed dxdt: 2 pk_sub + 1 pk_fma per pair.
    #define DXDT(Kd, Sv)                                                    \
        _Pragma("unroll")                                                   \
        for (int j = 0; j < NH; ++j)                                        \
            Kd[j] = __builtin_elementwise_fma(sh(Sv, j, 1) - sh(Sv, j, -2), \
                                              sh(Sv, j, -1), C8 - Sv[j]);

    // k1 = f(x)
    DXDT(K, X)
    #pragma unroll
    for (int j = 0; j < NH; ++j) A[j] = __builtin_elementwise_fma(C6, K[j], X[j]);
    #pragma unroll
    for (int j = 0; j < NH; ++j) S[j] = __builtin_elementwise_fma(H2, K[j], X[j]);

    // k2 = f(x + dt/2 k1)
    DXDT(K, S)
    #pragma unroll
    for (int j = 0; j < NH; ++j) A[j] = __builtin_elementwise_fma(C3, K[j], A[j]);
    #pragma unroll
    for (int j = 0; j < NH; ++j) S[j] = __builtin_elementwise_fma(H2, K[j], X[j]);

    // k3 = f(x + dt/2 k2)
    DXDT(K, S)
    #pragma unroll
    for (int j = 0; j < NH; ++j) A[j] = __builtin_elementwise_fma(C3, K[j], A[j]);
    #pragma unroll
    for (int j = 0; j < NH; ++j) S[j] = __builtin_elementwise_fma(DT, K[j], X[j]);

    // k4 = f(x + dt k3)
    DXDT(K, S)
    #pragma unroll
    for (int j = 0; j < NH; ++j) A[j] = __builtin_elementwise_fma(C6, K[j], A[j]);
    #undef DXDT

    // ---- Stage 3: un-interleave, write own row back (10x ds_store_b128) ----
    float as[NS];
    #pragma unroll
    for (int j = 0; j < NH; ++j) { as[j] = A[j].x; as[j + NH] = A[j].y; }

    #pragma unroll
    for (int j = 0; j < NS / 4; ++j) {
        float4 v;
        v.x = as[4 * j + 0]; v.y = as[4 * j + 1];
        v.z = as[4 * j + 2]; v.w = as[4 * j + 3];
        tile4[tid * (NS / 4) + j] = v;
    }
    __syncthreads();   // all rows visible before cross-thread async store

    // ---- Stage 4: coalesced async store LDS -> global ----
    #pragma unroll
    for (int it = 0; it < NIT; ++it) {
        asm volatile("global_store_async_from_lds_b128 %0, %1, %2 offset:%3"
                     :: "v"(voff), "v"(loff), "s"(gout), "i"(it * RPB * 16)
                     : "memory");
    }
    asm volatile("s_wait_asynccnt 0" ::: "memory");
}

extern "C" void kernel_launch(void* const* d_in, const int* in_sizes, int n_in,
                              void* d_out, int out_size, void* d_ws, size_t ws_size,
                              hipStream_t stream) {
    (void)n_in; (void)d_ws; (void)ws_size; (void)out_size;
    const float* x0  = (const float*)d_in[0];
    // d_in[1] = t (unused, autonomous system)
    const float* dtp = (const float*)d_in[2];
    float* out = (float*)d_out;

    const int batch  = in_sizes[0] / NS;     // 524288
    const int blocks = batch / RPB;          // 2048 (batch is a multiple of 256)

    lorenz96_rk4_kernel<<<blocks, RPB, 0, stream>>>(x0, dtp, out);
}